// topo_CNN_695784702495
// MI455X (gfx1250) — compile-verified
//
#include <hip/hip_runtime.h>
#include <hip/hip_bf16.h>

typedef __attribute__((ext_vector_type(2))) float v2f;
typedef __attribute__((ext_vector_type(8))) float v8f;

// Generic Conv1d(K=5,pad=2) -> ReLU -> MaxPool(2) block (VALU path, small stages).
// in: CIN x LIN, w: COUT x CIN x 5, out: COUT x (LIN/2)
template <int CIN, int LIN, int COUT>
__device__ __forceinline__ void conv_block(const float* __restrict__ in,
                                           const float* __restrict__ w,
                                           const float* __restrict__ bias,
                                           float* __restrict__ out,
                                           int tid, int nthr) {
    constexpr int LOUT = LIN / 2;
    for (int idx = tid; idx < COUT * LOUT; idx += nthr) {
        const int co = idx / LOUT;
        const int p  = idx - co * LOUT;
        const int l0 = 2 * p;
        float v0 = bias[co];
        float v1 = v0;
        for (int ci = 0; ci < CIN; ++ci) {
            const float* yr = in + ci * LIN;
            const float* wr = w + (co * CIN + ci) * 5;
            float t[6];
#pragma unroll
            for (int u = 0; u < 6; ++u) {
                const int l = l0 - 2 + u;
                t[u] = (l >= 0 && l < LIN) ? yr[l] : 0.0f;
            }
#pragma unroll
            for (int u = 0; u < 5; ++u) {
                const float ww = wr[u];
                v0 = fmaf(ww, t[u],     v0);
                v1 = fmaf(ww, t[u + 1], v1);
            }
        }
        // ReLU then MaxPool == max(v0,v1) clamped at 0
        out[idx] = fmaxf(fmaxf(v0, v1), 0.0f);
    }
}

__global__ __launch_bounds__(1024, 1) void topo_cnn_fused(
    const float* __restrict__ x,       // (64,1024)
    const float* __restrict__ topo,    // (64,64)
    const float* __restrict__ W1,      // (64,128,64)
    const float* __restrict__ W2,      // (64,64,128)
    const float* __restrict__ c1_w, const float* __restrict__ c1_b,
    const float* __restrict__ c2_w, const float* __restrict__ c2_b,
    const float* __restrict__ c3_w, const float* __restrict__ c3_b,
    const float* __restrict__ c4_w, const float* __restrict__ c4_b,
    const float* __restrict__ c5_w, const float* __restrict__ c5_b,
    const float* __restrict__ c6_w, const float* __restrict__ c6_b,
    const float* __restrict__ out_w, const float* __restrict__ out_b,
    float* __restrict__ d_out,
    float* __restrict__ y_ws)          // (64,1024) scratch in global ws
{
    // LDS layout (overlaid lifetimes): 26000 floats = 104 KB (<< 320 KB WGP LDS)
    //  s_xmean  [64]
    //  s_big    [16384]  reused: stage1 partials(1024) -> h(8192) -> p1(32x512)
    //  s_attn   [4096]
    //  s_p2     [4096] (16x256)  s_p3 [1024] (8x128)  s_p4 [256] (4x64)
    //  s_p5     [64]   (2x32)    s_p6 [16]   (1x16)
    __shared__ __align__(16) float smem[26000];
    float* s_xmean = smem;
    float* s_big   = smem + 64;
    float* s_attn  = s_big + 16384;
    float* s_p2    = s_attn + 4096;
    float* s_p3    = s_p2 + 4096;
    float* s_p4    = s_p3 + 1024;
    float* s_p5    = s_p4 + 256;
    float* s_p6    = s_p5 + 64;

    const int tid  = threadIdx.x;
    const int wave = tid >> 5;
    const int lane = tid & 31;
    const int half = lane >> 4;               // 0: lanes 0-15, 1: lanes 16-31
    const int l16  = lane & 15;

    // ---- Stage 1: xmean[c] = mean_l x[c][l] (deterministic tree reduce) ----
    {
        const int c = tid >> 4, j = tid & 15;
        const float* xr = x + c * 1024 + j;
        float s = 0.0f;
#pragma unroll 8
        for (int i = 0; i < 64; ++i) s += xr[i * 16];
        s_big[tid] = s;                       // partial[c*16 + j]
    }
    __syncthreads();
    if (tid < 64) {
        float s = 0.0f;
#pragma unroll
        for (int j = 0; j < 16; ++j) s += s_big[tid * 16 + j];
        s_xmean[tid] = s * (1.0f / 1024.0f);
    }
    __syncthreads();

    // ---- Stage 2a: h[i][j] = relu( sum_k xmean[k]*topo[i][k]*W1[i][j][k] ) ----
    for (int it = 0; it < 8; ++it) {
        const int idx = tid + it * 1024;      // < 8192
        const int i = idx >> 7, j = idx & 127;
        const float* w1r = W1 + (i * 128 + j) * 64;
        const float* tr  = topo + i * 64;
        float acc = 0.0f;
#pragma unroll 8
        for (int k = 0; k < 64; ++k)
            acc = fmaf(s_xmean[k] * tr[k], w1r[k], acc);
        s_big[idx] = fmaxf(acc, 0.0f);        // h
    }
    __syncthreads();

    // ---- Stage 2b: attn[i][o] = sigmoid( sum_h h[i][h]*W2[i][o][h] ) ----
    for (int it = 0; it < 4; ++it) {
        const int idx = tid + it * 1024;      // < 4096
        const int i = idx >> 6, o = idx & 63;
        const float* w2r = W2 + (i * 64 + o) * 128;
        const float* hr  = s_big + i * 128;
        float acc = 0.0f;
#pragma unroll 8
        for (int k = 0; k < 128; ++k)
            acc = fmaf(hr[k], w2r[k], acc);
        s_attn[idx] = 1.0f / (1.0f + __expf(-acc));
    }
    __syncthreads();

    // ---- Stage 3: y = attn(64x64) @ x(64x1024) via V_WMMA_F32_16X16X4_F32 ----
    // 4 x 64 tiles of 16x16; 32 waves * 8 tiles each; K=64 in 16 steps of 4.
    for (int t8 = 0; t8 < 8; ++t8) {
        const int tile = wave + 32 * t8;      // 0..255
        const int m0 = (tile >> 6) * 16;      // row tile (0..3)*16
        const int n0 = (tile & 63) * 16;      // col tile (0..63)*16
        v8f c = {};
#pragma unroll
        for (int kk = 0; kk < 16; ++kk) {
            const int k0 = kk * 4 + 2 * half;
            // A fragment: attn[M=m0+l16][K = k0, k0+1] per ISA 16x4 f32 layout
            v2f a, b;
            a.x = s_attn[(m0 + l16) * 64 + k0];
            a.y = s_attn[(m0 + l16) * 64 + k0 + 1];
            // B fragment: x[K][N=n0+l16], rows striped across lanes per VGPR
            b.x = x[(k0)     * 1024 + n0 + l16];
            b.y = x[(k0 + 1) * 1024 + n0 + l16];
            c = __builtin_amdgcn_wmma_f32_16x16x4_f32(
                    false, a, false, b, (short)0, c, false, false);
        }
        // D layout: VGPR j -> row m0 + j + 8*half, col n0 + l16
#pragma unroll
        for (int j = 0; j < 8; ++j)
            y_ws[(m0 + j + 8 * half) * 1024 + n0 + l16] = c[j];
    }
    __threadfence();   // y_ws (global) must be visible to this WG's readers
    __syncthreads();

    // ---- Stage 4a: conv1 on the matrix pipe ----
    // z(32x1024) = sum_{t=0..4} W_t(32x64) @ y_shift_t(64x1024)
    // (tap-decomposed GEMM; no im2col materialization). Bias + ReLU +
    // MaxPool(2) fused in the epilogue via lane-pair exchange.
    // 2 x 64 tiles of 16x16; 32 waves * 4 tiles; K=5*64 in 80 WMMA steps.
    {
        for (int tt = 0; tt < 4; ++tt) {
            const int tile = wave + 32 * tt;  // 0..127
            const int m0 = (tile >> 6) * 16;  // 0 or 16
            const int n0 = (tile & 63) * 16;
            const int col = n0 + l16;
            v8f c = {};
            for (int t = 0; t < 5; ++t) {
                const int src   = col + t - 2;
                const int srcc  = min(max(src, 0), 1023);   // clamped (safe addr)
                const bool okay = (src >= 0) && (src < 1024);
#pragma unroll
                for (int kk = 0; kk < 16; ++kk) {
                    const int k0 = kk * 4 + 2 * half;
                    v2f a, b;
                    // A: W_t[M=m0+l16][K=ic] = c1_w[co][ic][t]
                    a.x = c1_w[((m0 + l16) * 64 + k0)     * 5 + t];
                    a.y = c1_w[((m0 + l16) * 64 + k0 + 1) * 5 + t];
                    // B: y[ic=K][col + t - 2], zero outside [0,1024)
                    b.x = okay ? y_ws[(k0)     * 1024 + srcc] : 0.0f;
                    b.y = okay ? y_ws[(k0 + 1) * 1024 + srcc] : 0.0f;
                    c = __builtin_amdgcn_wmma_f32_16x16x4_f32(
                            false, a, false, b, (short)0, c, false, false);
                }
            }
            // Epilogue: +bias, ReLU, horizontal MaxPool(2). Adjacent columns
            // live in lane pairs (lane^1 within each half) -> one shuffle.
#pragma unroll
            for (int j = 0; j < 8; ++j) {
                const int row = m0 + j + 8 * half;
                const float v = c[j] + c1_b[row];
                const float m = fmaxf(v, __shfl_xor(v, 1, 32));
                if ((l16 & 1) == 0)
                    s_big[row * 512 + (col >> 1)] = fmaxf(m, 0.0f);  // p1: 32x512
            }
        }
    }
    __syncthreads();

    // ---- Stage 4b: remaining conv chain (tiny; VALU) ----
    conv_block<32, 512, 16>(s_big, c2_w, c2_b, s_p2, tid, 1024);   // 16x256
    __syncthreads();
    conv_block<16, 256, 8>(s_p2, c3_w, c3_b, s_p3, tid, 1024);     // 8x128
    __syncthreads();
    conv_block<8, 128, 4>(s_p3, c4_w, c4_b, s_p4, tid, 1024);      // 4x64
    __syncthreads();
    conv_block<4, 64, 2>(s_p4, c5_w, c5_b, s_p5, tid, 1024);       // 2x32
    __syncthreads();
    conv_block<2, 32, 1>(s_p5, c6_w, c6_b, s_p6, tid, 1024);       // 1x16
    __syncthreads();

    // ---- Stage 5: out = p6 . out_w + out_b ----
    if (tid == 0) {
        float acc = out_b[0];
#pragma unroll
        for (int k = 0; k < 16; ++k)
            acc = fmaf(s_p6[k], out_w[k], acc);
        d_out[0] = acc;
    }
}

extern "C" void kernel_launch(void* const* d_in, const int* in_sizes, int n_in,
                              void* d_out, int out_size, void* d_ws, size_t ws_size,
                              hipStream_t stream) {
    (void)in_sizes; (void)n_in; (void)out_size; (void)ws_size;
    const float* x     = (const float*)d_in[0];
    const float* topo  = (const float*)d_in[1];
    const float* W1    = (const float*)d_in[2];
    const float* W2    = (const float*)d_in[3];
    const float* c1_w  = (const float*)d_in[4];
    const float* c1_b  = (const float*)d_in[5];
    const float* c2_w  = (const float*)d_in[6];
    const float* c2_b  = (const float*)d_in[7];
    const float* c3_w  = (const float*)d_in[8];
    const float* c3_b  = (const float*)d_in[9];
    const float* c4_w  = (const float*)d_in[10];
    const float* c4_b  = (const float*)d_in[11];
    const float* c5_w  = (const float*)d_in[12];
    const float* c5_b  = (const float*)d_in[13];
    const float* c6_w  = (const float*)d_in[14];
    const float* c6_b  = (const float*)d_in[15];
    const float* out_w = (const float*)d_in[16];
    const float* out_b = (const float*)d_in[17];

    float* y_ws = (float*)d_ws;   // 64*1024 floats = 256 KB scratch for attn@x

    topo_cnn_fused<<<dim3(1), dim3(1024), 0, stream>>>(
        x, topo, W1, W2, c1_w, c1_b, c2_w, c2_b, c3_w, c3_b,
        c4_w, c4_b, c5_w, c5_b, c6_w, c6_b, out_w, out_b,
        (float*)d_out, y_ws);
}